// RGLRU_71347996721888
// MI455X (gfx1250) — compile-verified
//
#include <hip/hip_runtime.h>
#include <math.h>

// ---------------- problem constants ----------------
#define WIDTH     2048
#define NUM_HEADS 16
#define HEAD_DIM  128          // WIDTH / NUM_HEADS
#define BATCH     16384
#define ROWS      128          // rows of x per workgroup
#define XS_STRIDE 132          // HEAD_DIM + 4 pad (A-frag bank spread)
#define WP_STRIDE 288          // 2*HEAD_DIM + 32 pad (pair-major row; 288 % 64 == 32)
#define KPAIRS    (HEAD_DIM/2) // 64

typedef float v2f __attribute__((ext_vector_type(2)));
typedef float v4f __attribute__((ext_vector_type(4)));
typedef float v8f __attribute__((ext_vector_type(8)));

__global__ __launch_bounds__(256) void rglru_fused_kernel(
    const float* __restrict__ x, const float* __restrict__ state,
    const float* __restrict__ w_in, const float* __restrict__ b_in,
    const float* __restrict__ w_a,  const float* __restrict__ b_a,
    const float* __restrict__ a_param, float* __restrict__ out)
{
    extern __shared__ float lds[];
    float* Xs   = lds;                          // [ROWS][XS_STRIDE] row-major x tile
    float* Wpin = lds + ROWS * XS_STRIDE;       // [KPAIRS][WP_STRIDE] pair-major w_in[h]
    float* Wpa  = Wpin + KPAIRS * WP_STRIDE;    // pair-major w_a[h]

    const int head = blockIdx.y;
    const int row0 = blockIdx.x * ROWS;
    const int tid  = threadIdx.x;

    // ---- stage x tile: coalesced b128 global loads -> LDS b128 stores ----
    {
        const float* xg = x + (size_t)row0 * WIDTH + (size_t)head * HEAD_DIM;
        const int r = tid >> 5;         // 0..7
        const int c = (tid & 31) * 4;   // 0..124
        #pragma unroll
        for (int i = 0; i < ROWS / 8; ++i) {
            v4f v = __builtin_nontemporal_load(
                (const v4f*)(xg + (size_t)(r + 8 * i) * WIDTH + c));
            *(v4f*)&Xs[(r + 8 * i) * XS_STRIDE + c] = v;
        }
    }
    // ---- stage weights pair-major: Wp[k/2][2n + (k&1)] = w[k][n] ----
    {
        const float* wi = w_in + (size_t)head * HEAD_DIM * HEAD_DIM;
        const float* wa = w_a  + (size_t)head * HEAD_DIM * HEAD_DIM;
        #pragma unroll
        for (int i = 0; i < (KPAIRS * 64) / 256; ++i) {  // 16 iters
            int idx = i * 256 + tid;
            int kp  = idx >> 6;          // 0..63
            int n   = (idx & 63) * 2;    // 0..126 (even)
            v2f a0 = *(const v2f*)(wi + (size_t)(2 * kp)     * HEAD_DIM + n);
            v2f a1 = *(const v2f*)(wi + (size_t)(2 * kp + 1) * HEAD_DIM + n);
            *(v4f*)&Wpin[kp * WP_STRIDE + 2 * n] = (v4f){a0.x, a1.x, a0.y, a1.y};
            v2f c0 = *(const v2f*)(wa + (size_t)(2 * kp)     * HEAD_DIM + n);
            v2f c1 = *(const v2f*)(wa + (size_t)(2 * kp + 1) * HEAD_DIM + n);
            *(v4f*)&Wpa[kp * WP_STRIDE + 2 * n] = (v4f){c0.x, c1.x, c0.y, c1.y};
        }
    }
    __syncthreads();

    const int wave = tid >> 5;     // 0..7 (wave32)
    const int lane = tid & 31;
    const int half = lane >> 4;    // hi/lo 16-lane group
    const int l15  = lane & 15;
    const int m0   = wave * 16;    // this wave's 16-row strip

    // per-output-column values for this lane (N = t*16 + l15)
    float nsp8_v[8];               // -8 * softplus(a_param[n])
    v8f accIn[8], accA[8];
    #pragma unroll
    for (int t = 0; t < 8; ++t) {
        int n = t * 16 + l15;
        float bi = b_in[head * HEAD_DIM + n];
        float ba = b_a [head * HEAD_DIM + n];
        float ap = a_param[head * HEAD_DIM + n];
        float z  = fabsf(ap);                       // stable softplus (precise path, 8/lane)
        nsp8_v[t] = -8.0f * (fmaxf(ap, 0.0f) + log1pf(expf(-z)));
        v8f ci, ca;
        #pragma unroll
        for (int r = 0; r < 8; ++r) { ci[r] = bi; ca[r] = ba; }
        accIn[t] = ci;  accA[t] = ca;               // bias folded into C
    }

    // A(16x4) frag: lane -> (M = l15, K = kk + 2*half + v), v2f contiguous in k
    const float* aRowBase = &Xs[(m0 + l15) * XS_STRIDE + 2 * half];
    // B(4x16) frag: lane -> (K = kk + 2*half + v, N = t*16 + l15); pair-major => one b64
    const float* bBaseIn  = &Wpin[half * WP_STRIDE + 2 * l15];
    const float* bBaseA   = &Wpa [half * WP_STRIDE + 2 * l15];

    for (int kk = 0; kk < HEAD_DIM; kk += 4) {
        v2f a = *(const v2f*)(aRowBase + kk);
        const int kpOff = (kk >> 1) * WP_STRIDE;
        #pragma unroll
        for (int t = 0; t < 8; ++t) {
            v2f bi = *(const v2f*)(bBaseIn + kpOff + t * 32);
            accIn[t] = __builtin_amdgcn_wmma_f32_16x16x4_f32(
                false, a, false, bi, (short)0, accIn[t], false, false);
            v2f bb = *(const v2f*)(bBaseA + kpOff + t * 32);
            accA[t] = __builtin_amdgcn_wmma_f32_16x16x4_f32(
                false, a, false, bb, (short)0, accA[t], false, false);
        }
    }

    // ---- fused epilogue: C/D layout (M = r + 8*half, N = t*16 + l15) ----
    // Fast-math VALU: v_exp_f32 / v_rcp_f32 / v_sqrt_f32 keep the epilogue from
    // stealing issue slots from the WMMA pipe (few-ulp error, fine vs fp32 ref).
    const size_t colBase = (size_t)head * HEAD_DIM;
    #pragma unroll
    for (int t = 0; t < 8; ++t) {
        const int n = t * 16 + l15;
        #pragma unroll
        for (int r = 0; r < 8; ++r) {
            const int   mloc = m0 + r + 8 * half;
            const size_t off = (size_t)(row0 + mloc) * WIDTH + colBase + n;
            float yin = accIn[t][r];
            float ya  = accA[t][r];
            float gx  = __builtin_amdgcn_rcpf(1.0f + __expf(-yin));
            float ga  = __builtin_amdgcn_rcpf(1.0f + __expf(-ya));
            float la  = ga * nsp8_v[t];
            float av  = __expf(la);
            float sc  = __builtin_amdgcn_sqrtf(fmaxf(1.0f - av * av, 0.0f));
            float xv  = Xs[mloc * XS_STRIDE + n];
            float sv  = __builtin_nontemporal_load(state + off);
            __builtin_nontemporal_store(av * sv + gx * xv * sc, out + off);
        }
    }
}

extern "C" void kernel_launch(void* const* d_in, const int* in_sizes, int n_in,
                              void* d_out, int out_size, void* d_ws, size_t ws_size,
                              hipStream_t stream) {
    const float* x       = (const float*)d_in[0];
    const float* state   = (const float*)d_in[1];
    const float* w_in    = (const float*)d_in[2];
    const float* b_in    = (const float*)d_in[3];
    const float* w_a     = (const float*)d_in[4];
    const float* b_a     = (const float*)d_in[5];
    const float* a_param = (const float*)d_in[6];
    float* out = (float*)d_out;

    const size_t shmem =
        (size_t)(ROWS * XS_STRIDE + 2 * KPAIRS * WP_STRIDE) * sizeof(float); // 215,040 B

    // Allow >default dynamic LDS (WGP has 320 KB); not a stream op, capture-safe.
    (void)hipFuncSetAttribute((const void*)rglru_fused_kernel,
                              hipFuncAttributeMaxDynamicSharedMemorySize, (int)shmem);

    dim3 grid(BATCH / ROWS, NUM_HEADS);  // 128 x 16 = 2048 workgroups
    rglru_fused_kernel<<<grid, 256, shmem, stream>>>(
        x, state, w_in, b_in, w_a, b_a, a_param, out);
}